// _QuantumLSTMCell_65481071404662
// MI455X (gfx1250) — compile-verified
//
#include <hip/hip_runtime.h>

typedef __attribute__((ext_vector_type(2))) float v2f;
typedef __attribute__((ext_vector_type(8))) float v8f;

__device__ __forceinline__ float fsig(float z) {
    return 1.0f / (1.0f + __expf(-z));
}
__device__ __forceinline__ float ftanh(float z) {
    float e = __expf(-2.0f * z);
    return (1.0f - e) / (1.0f + e);
}

// One wave computes a 16-row tile of angles = comb @ W_all^T  (16 output cols =
// 4 gates x 4 wires) with V_WMMA_F32_16X16X4_F32, then the closed-form quantum
// epilogue:  <Z0>=c1c2c3, <Z1>=c0c1, <Z2>=c0c1c2, <Z3>=c0c1c2c3, c_w=cos(A_w).
__global__ __launch_bounds__(256)
void qlstm_kernel(const float* __restrict__ x,
                  const float* __restrict__ hx,
                  const float* __restrict__ cx,
                  const float* __restrict__ W_f, const float* __restrict__ b_f,
                  const float* __restrict__ W_i, const float* __restrict__ b_i,
                  const float* __restrict__ W_u, const float* __restrict__ b_u,
                  const float* __restrict__ W_o, const float* __restrict__ b_o,
                  const float* __restrict__ th_f, const float* __restrict__ th_i,
                  const float* __restrict__ th_u, const float* __restrict__ th_o,
                  float* __restrict__ h_out, float* __restrict__ c_out)
{
    __shared__ float sh[8][2][16][4];   // [wave][h|c][row][wire]

    const int lane  = threadIdx.x & 31;
    const int wid   = threadIdx.x >> 5;
    const long gw   = (long)blockIdx.x * 8 + wid;
    const long rowbase = gw * 16;

    const int hi16 = lane >> 4;          // half-wave select
    const int n    = lane & 15;          // output column / A-row index
    const int gate = n >> 2;             // 0:f 1:i 2:u 3:o
    const int wire = n & 3;
    const int koff = hi16 * 2;           // K offset inside the 4-wide step

    // A operand: row (rowbase+n) of comb = [x | hx]
    const long rowA = rowbase + n;
    const float* xrow  = x  + rowA * 256 + koff;
    const float* hxrow = hx + rowA * 4   + koff;

    // B operand: weight row (wire) of this lane's gate
    const float* Wg = (gate == 0) ? W_f : (gate == 1) ? W_i
                    : (gate == 2) ? W_u : W_o;
    const float* Wrow = Wg + wire * 260 + koff;

    const float* bg = (gate == 0) ? b_f : (gate == 1) ? b_i
                    : (gate == 2) ? b_u : b_o;
    const float* tg = (gate == 0) ? th_f : (gate == 1) ? th_i
                    : (gate == 2) ? th_u : th_o;
    const float bt = bg[wire] + tg[wire];   // bias + trainable RX angle

    v8f acc = {0.f, 0.f, 0.f, 0.f, 0.f, 0.f, 0.f, 0.f};

    // 64 K-steps over x (K = 0..255)
    #pragma unroll 8
    for (int k = 0; k < 256; k += 4) {
        v2f a = *(const v2f*)(xrow + k);
        v2f b = *(const v2f*)(Wrow + k);
        acc = __builtin_amdgcn_wmma_f32_16x16x4_f32(
                  false, a, false, b, (short)0, acc, false, false);
    }
    // tail K-step from hx (K = 256..259)
    {
        v2f a = *(const v2f*)(hxrow);
        v2f b = *(const v2f*)(Wrow + 256);
        acc = __builtin_amdgcn_wmma_f32_16x16x4_f32(
                  false, a, false, b, (short)0, acc, false, false);
    }

    const int qbase = lane & ~3;   // quad = the 4 wires of this lane's gate
    const int hbase = lane & 16;   // half-wave base for cross-gate gather
    const int rbase = hi16 * 8;    // C-layout: half 0 holds M=0..7, half 1 M=8..15

    // Branchless per-lane selection constants (hoisted out of the r-loop):
    //   z = sel0 * c1 * sel2 * sel3   (verified for all four wires)
    const bool use_c0 = (wire != 0);
    const bool use_c2 = (wire != 1);
    const bool use_c3 = (wire == 0) || (wire == 3);
    // activation: act = fma(sigmoid(m*z), m, a); tanh(z) = 2*sigmoid(2z)-1
    const float m = (gate == 2) ? 2.0f : 1.0f;
    const float aoff = (gate == 2) ? -1.0f : 0.0f;

    #pragma unroll
    for (int r = 0; r < 8; ++r) {
        float ang = acc[r] + bt;
        float cw  = __cosf(ang);
        float c0 = __shfl(cw, qbase + 0, 32);
        float c1 = __shfl(cw, qbase + 1, 32);
        float c2 = __shfl(cw, qbase + 2, 32);
        float c3 = __shfl(cw, qbase + 3, 32);
        float z = (use_c0 ? c0 : 1.0f) * c1
                * (use_c2 ? c2 : 1.0f)
                * (use_c3 ? c3 : 1.0f);
        float act = __builtin_fmaf(fsig(m * z), m, aoff);
        // gather f/i/g/o for this lane's wire from lanes w, w+4, w+8, w+12
        float fv = __shfl(act, hbase + wire + 0,  32);
        float iv = __shfl(act, hbase + wire + 4,  32);
        float gv = __shfl(act, hbase + wire + 8,  32);
        float ov = __shfl(act, hbase + wire + 12, 32);
        if (gate == 0) {   // lanes 0..3 / 16..19 are the writers (wire == n)
            float cxv = cx[(rowbase + rbase + r) * 4 + wire];
            float cn  = fv * cxv + iv * gv;
            float hn  = ov * ftanh(cn);
            sh[wid][0][rbase + r][wire] = hn;
            sh[wid][1][rbase + r][wire] = cn;
        }
    }
    __syncthreads();

    // Coalesced float4 stores: lanes 0-15 store h rows, lanes 16-31 store c rows
    if (hi16 == 0) {
        float4 hv = *(const float4*)&sh[wid][0][n][0];
        *(float4*)(h_out + (rowbase + n) * 4) = hv;
    } else {
        float4 cv = *(const float4*)&sh[wid][1][n][0];
        *(float4*)(c_out + (rowbase + n) * 4) = cv;
    }
}

extern "C" void kernel_launch(void* const* d_in, const int* in_sizes, int n_in,
                              void* d_out, int out_size, void* d_ws, size_t ws_size,
                              hipStream_t stream) {
    const float* x    = (const float*)d_in[0];
    const float* hx   = (const float*)d_in[1];
    const float* cx   = (const float*)d_in[2];
    const float* W_f  = (const float*)d_in[3];
    const float* b_f  = (const float*)d_in[4];
    const float* W_i  = (const float*)d_in[5];
    const float* b_i  = (const float*)d_in[6];
    const float* W_u  = (const float*)d_in[7];
    const float* b_u  = (const float*)d_in[8];
    const float* W_o  = (const float*)d_in[9];
    const float* b_o  = (const float*)d_in[10];
    const float* th_f = (const float*)d_in[11];
    const float* th_i = (const float*)d_in[12];
    const float* th_u = (const float*)d_in[13];
    const float* th_o = (const float*)d_in[14];

    const int B = in_sizes[0] / 256;          // 131072 rows
    float* h_out = (float*)d_out;             // [B,4]
    float* c_out = h_out + (size_t)B * 4;     // [B,4]

    dim3 block(256);                          // 8 waves -> 128 rows per block
    dim3 grid(B / 128);                       // 1024 blocks, no partial tiles
    hipLaunchKernelGGL(qlstm_kernel, grid, block, 0, stream,
                       x, hx, cx, W_f, b_f, W_i, b_i, W_u, b_u, W_o, b_o,
                       th_f, th_i, th_u, th_o, h_out, c_out);
}